// GaussianSplattingCompliance_22419729285764
// MI455X (gfx1250) — compile-verified
//
#include <hip/hip_runtime.h>
#include <math.h>

// ---------------------------------------------------------------------------
// GaussianSplattingCompliance for MI455X (gfx1250, wave32).
//
// z[b,n] = quadratic(x_n, y_n; gaussian b) is factored into a GEMM:
//   z' = (log2e * Coef)[64x8] * Feat[8xN],  Feat = [x^2, y^2, xy, x, y, 1, 0, 0]
// evaluated with V_WMMA_F32_16X16X4_F32 (two chained K=4 WMMAs => K=8),
// 4 gaussian tiles of 16 => all 64 gaussians per 16-point column tile.
// kernel = exp2(z') directly (log2e folded into coefficients -> no mul).
// H_splitted sigmoid uses V_TANH_F32 (1 trans op) instead of exp+rcp (2).
// ---------------------------------------------------------------------------

typedef __attribute__((ext_vector_type(2))) float v2f;
typedef __attribute__((ext_vector_type(8))) float v8f;

#define THREADS 256
#define NBLK    400

#if __has_builtin(__builtin_amdgcn_tanhf)
#define USE_TANH 1
#else
#define USE_TANH 0
#endif

// workspace layout (32-bit words):
// [0..3]    encoded min_x, min_y, max_x, max_y (ordered-uint)
// [8..519]  coef[64][8]  (pre-scaled by log2e)
// [520..]   partials[NBLK][3]
#define WS_COEF 8
#define WS_PART 520

// monotonic float<->uint encoding: unsigned compare == float compare
__device__ __forceinline__ unsigned encf(float f) {
    unsigned b = __float_as_uint(f);
    return (b & 0x80000000u) ? ~b : (b | 0x80000000u);
}
__device__ __forceinline__ float decf(unsigned u) {
    unsigned b = (u & 0x80000000u) ? (u ^ 0x80000000u) : ~u;
    return __uint_as_float(b);
}
__device__ __forceinline__ float sigf(float x) { return 1.0f / (1.0f + expf(-x)); }

__global__ void k_init(unsigned* __restrict__ wsu) {
    int t = threadIdx.x;
    if (t < 2)      wsu[t] = 0xFFFFFFFFu;   // min accumulators
    else if (t < 4) wsu[t] = 0u;            // max accumulators
}

__global__ void k_minmax(const float* __restrict__ coords, int n, unsigned* __restrict__ wsu) {
    float mnx = 3.402823466e38f, mny = 3.402823466e38f;
    float mxx = -3.402823466e38f, mxy = -3.402823466e38f;
    for (int i = blockIdx.x * blockDim.x + threadIdx.x; i < n; i += gridDim.x * blockDim.x) {
        float x = coords[2 * i], y = coords[2 * i + 1];
        mnx = fminf(mnx, x); mny = fminf(mny, y);
        mxx = fmaxf(mxx, x); mxy = fmaxf(mxy, y);
    }
    __shared__ float s0[THREADS], s1[THREADS], s2[THREADS], s3[THREADS];
    int tid = threadIdx.x;
    s0[tid] = mnx; s1[tid] = mny; s2[tid] = mxx; s3[tid] = mxy;
    __syncthreads();
    for (int off = THREADS / 2; off > 0; off >>= 1) {
        if (tid < off) {
            s0[tid] = fminf(s0[tid], s0[tid + off]);
            s1[tid] = fminf(s1[tid], s1[tid + off]);
            s2[tid] = fmaxf(s2[tid], s2[tid + off]);
            s3[tid] = fmaxf(s3[tid], s3[tid + off]);
        }
        __syncthreads();
    }
    if (tid == 0) {
        atomicMin(&wsu[0], encf(s0[0]));
        atomicMin(&wsu[1], encf(s1[0]));
        atomicMax(&wsu[2], encf(s2[0]));
        atomicMax(&wsu[3], encf(s3[0]));
    }
}

// Per-gaussian quadratic coefficients (pre-scaled by log2e so the main loop
// can use exp2 directly): z*log2e = P x^2 + Q y^2 + R xy + cx x + cy y + c0
__global__ void k_params(const float* __restrict__ Ws, const float* __restrict__ Wr,
                         const float* __restrict__ Wrot, const float* __restrict__ Woff,
                         const unsigned* __restrict__ wsu, float* __restrict__ coef) {
    int b = threadIdx.x;
    if (b >= 64) return;
    const float SCALE_SIGMA = 0.06f / 4.0f;
    const float SCALE_MAX = SCALE_SIGMA * 2.0f;
    const float SCALE_MIN = SCALE_SIGMA * 0.75f;
    const float PI  = 3.14159265358979323846f;
    const float L2E = 1.44269504088896340736f;   // log2(e)

    float mnx = decf(wsu[0]), mny = decf(wsu[1]);
    float mxx = decf(wsu[2]), mxy = decf(wsu[3]);

    float bs    = (SCALE_MAX - SCALE_MIN) * sigf(Ws[b]) + SCALE_MIN + 1e-8f;
    float ratio = 2.0f * sigf(Wr[b]) + 0.5f;                 // RATIO in [0.5, 2.5]
    float rot   = -0.5f * PI + PI * sigf(Wrot[b]);
    float ox    = mnx + (mxx - mnx) * sigf(Woff[2 * b]);
    float oy    = mny + (mxy - mny) * sigf(Woff[2 * b + 1]);

    float c = cosf(rot), s = sinf(rot);
    float icd0 = 1.0f / (1.0f + 1e-6f);
    float icd1 = 1.0f / (ratio * ratio + 1e-6f);
    float ib2  = 1.0f / (bs * bs);

    float P  = -0.5f * ib2 * (icd0 * c * c + icd1 * s * s);
    float Q  = -0.5f * ib2 * (icd0 * s * s + icd1 * c * c);
    float R  = -ib2 * c * s * (icd1 - icd0);
    float cx = -2.0f * P * ox - R * oy;
    float cy = -2.0f * Q * oy - R * ox;
    float c0 = P * ox * ox + Q * oy * oy + R * ox * oy;

    float* row = coef + 8 * b;
    row[0] = L2E * P;  row[1] = L2E * Q;  row[2] = L2E * R;  row[3] = L2E * cx;
    row[4] = L2E * cy; row[5] = L2E * c0; row[6] = 0.0f;     row[7] = 0.0f;
}

__global__ void __launch_bounds__(THREADS)
k_main(const float* __restrict__ coords, const float* __restrict__ ce, int n,
       const float* __restrict__ coef, float* __restrict__ partials) {
    const int lane   = threadIdx.x & 31;
    const int hi     = lane >> 4;         // lane-half: 0 or 1
    const int col    = lane & 15;         // column within 16-point tile
    const int wid    = (blockIdx.x * blockDim.x + threadIdx.x) >> 5;
    const int nwaves = (gridDim.x * blockDim.x) >> 5;
    const int ntiles = (n + 15) >> 4;

    // A fragments: 16x4 f32 layout -> lanes 0-15: rows M, K={k0,k0+1};
    // lanes 16-31: rows M, K={k0+2,k0+3}. kp=0 covers K 0..3, kp=1 covers K 4..7.
    v2f a[4][2];
#pragma unroll
    for (int g = 0; g < 4; ++g)
#pragma unroll
        for (int kp = 0; kp < 2; ++kp) {
            int row = g * 16 + col;
            int k0  = kp * 4 + hi * 2;
            a[g][kp] = *(const v2f*)(coef + row * 8 + k0);
        }

    float comp = 0.0f, hsum = 0.0f, ovl = 0.0f;

    for (int t = wid; t < ntiles; t += nwaves) {
        int np = t * 16 + col;
        int nc = np < n ? np : 0;
        float x = coords[2 * nc], y = coords[2 * nc + 1];

        // B fragments (assumed layout: VGPR v -> row K=v in lanes 0-15, K=v+2 in 16-31)
        v2f b0, b1;
        b0.x = hi ? x * y : x * x;   // K2 | K0
        b0.y = hi ? x     : y * y;   // K3 | K1
        b1.x = hi ? 0.0f  : y;       // K6 | K4
        b1.y = hi ? 0.0f  : 1.0f;    // K7 | K5

        float ks = 0.0f, hacc = 0.0f;
#pragma unroll
        for (int g = 0; g < 4; ++g) {
            v8f acc = {0.f, 0.f, 0.f, 0.f, 0.f, 0.f, 0.f, 0.f};
            acc = __builtin_amdgcn_wmma_f32_16x16x4_f32(false, a[g][0], false, b0,
                                                        (short)0, acc, false, false);
            acc = __builtin_amdgcn_wmma_f32_16x16x4_f32(false, a[g][1], false, b1,
                                                        (short)0, acc, false, false);
#pragma unroll
            for (int e = 0; e < 8; ++e) {
                // acc holds z*log2e -> gaussian kernel value with a bare v_exp_f32
                float kv = __builtin_amdgcn_exp2f(acc[e]);
                ks += kv;
#if USE_TANH
                // sig(57.6*kv - 5.76) = 0.5 + 0.5*tanh(28.8*kv - 2.88); affine part
                // hoisted out of the sum (hss = 32 + 0.5*Sum(tanh)).
                hacc += __builtin_amdgcn_tanhf(fmaf(28.8f, kv, -2.88f));
#else
                hacc += 1.0f / (1.0f + __expf(5.76f - 57.6f * kv));
#endif
            }
        }
        // combine lane halves: each half holds 32 of the 64 gaussians per column
        ks   += __shfl_xor(ks, 16);
        hacc += __shfl_xor(hacc, 16);
        ks   += 1e-8f;
#if USE_TANH
        float hss = fmaf(0.5f, hacc, 32.0f);                       // 64 sigmoid terms
        float H   = fmaf(0.5f, __builtin_amdgcn_tanhf(fmaf(-16.0f, ks, 6.4f)), 0.5f);
#else
        float hss = hacc;
        float H   = (1.0f - 1e-9f) / (1.0f + __expf(32.0f * (ks - 0.4f))) + 1e-9f;
#endif
        float hc = fminf(fmaxf(H, 1e-9f), 1.0f);
        float hv = hc * hc * hc;                                   // H^PENAL, PENAL=3
        float m  = (hi == 0 && np < n) ? 1.0f : 0.0f;              // one half owns the point
        comp = fmaf(m * hv, ce[nc], comp);
        hsum = fmaf(m, H, hsum);
        ovl += m * fmaxf(hss - 1.5f, 0.0f);
    }

    __shared__ float sc[THREADS], sh[THREADS], so[THREADS];
    int tid = threadIdx.x;
    sc[tid] = comp; sh[tid] = hsum; so[tid] = ovl;
    __syncthreads();
    for (int off = THREADS / 2; off > 0; off >>= 1) {
        if (tid < off) {
            sc[tid] += sc[tid + off];
            sh[tid] += sh[tid + off];
            so[tid] += so[tid + off];
        }
        __syncthreads();
    }
    if (tid == 0) {
        partials[3 * blockIdx.x + 0] = sc[0];
        partials[3 * blockIdx.x + 1] = sh[0];
        partials[3 * blockIdx.x + 2] = so[0];
    }
}

__global__ void k_final(const float* __restrict__ partials, int nblk, int n,
                        float* __restrict__ out) {
    if (threadIdx.x != 0 || blockIdx.x != 0) return;
    float comp = 0.0f, hsum = 0.0f, ovl = 0.0f;
    for (int i = 0; i < nblk; ++i) {          // fixed order -> deterministic
        comp += partials[3 * i + 0];
        hsum += partials[3 * i + 1];
        ovl  += partials[3 * i + 2];
    }
    float invn    = 1.0f / (float)n;
    float volfrac = fmaxf(hsum * invn - 0.6f, 0.0f);
    float overlap = ovl * invn;
    out[0] = -comp + 10.0f * volfrac + overlap;   // obj_ce
    out[1] = volfrac;                             // volfrac_loss_pre
    out[2] = overlap;                             // gaussian_overlap
    out[3] = comp;                                // compliance
    out[4] = comp + 10.0f * volfrac + overlap;    // obj_real
}

extern "C" void kernel_launch(void* const* d_in, const int* in_sizes, int n_in,
                              void* d_out, int out_size, void* d_ws, size_t ws_size,
                              hipStream_t stream) {
    (void)n_in; (void)out_size; (void)ws_size;
    const float* Ws     = (const float*)d_in[0];   // W_scale        [64,1]
    const float* Wr     = (const float*)d_in[1];   // W_sigmas_ratio [64,1]
    const float* Wrot   = (const float*)d_in[2];   // W_rotation     [64,1]
    const float* Woff   = (const float*)d_in[3];   // W_offsets      [64,2]
    const float* coords = (const float*)d_in[4];   // coords         [N,2]
    const float* ce     = (const float*)d_in[5];   // ce             [N]
    int n = in_sizes[5];

    unsigned* wsu = (unsigned*)d_ws;
    float* wsf    = (float*)d_ws;
    float* coef   = wsf + WS_COEF;
    float* part   = wsf + WS_PART;
    float* out    = (float*)d_out;

    hipLaunchKernelGGL(k_init,   dim3(1),    dim3(64),      0, stream, wsu);
    hipLaunchKernelGGL(k_minmax, dim3(128),  dim3(THREADS), 0, stream, coords, n, wsu);
    hipLaunchKernelGGL(k_params, dim3(1),    dim3(64),      0, stream, Ws, Wr, Wrot, Woff, wsu, coef);
    hipLaunchKernelGGL(k_main,   dim3(NBLK), dim3(THREADS), 0, stream, coords, ce, n, coef, part);
    hipLaunchKernelGGL(k_final,  dim3(1),    dim3(1),       0, stream, part, NBLK, n, out);
}